// DecoderOnlyAttention_43654047596877
// MI455X (gfx1250) — compile-verified
//
#include <hip/hip_runtime.h>

typedef __attribute__((ext_vector_type(2))) float v2f;
typedef __attribute__((ext_vector_type(8))) float v8f;

#define NEG_BIG (-3.4028234663852886e38f)

// Paged-KV GQA decode attention (flash-decode style), fp32.
// Grid: B*KVH workgroups of 256 threads (8 wave32 waves).
// Each wave: 16-token tiles, QK^T via 4 independent chains of
// v_wmma_f32_16x16x4_f32, online softmax, float4 PV accumulation;
// next-tile K/V rows prefetched with global_prefetch_b8;
// waves merged via LDS log-sum-exp combine.
__global__ __launch_bounds__(256)
void DecoderOnlyAttention_43654047596877_kernel(
    const float* __restrict__ Q,      // [B,H,1,D]
    const float* __restrict__ Knew,   // [B,KVH,1,D]
    const float* __restrict__ Vnew,   // [B,KVH,1,D]
    const float* __restrict__ Cos,    // [B,1,1,D]
    const float* __restrict__ Sin,    // [B,1,1,D]
    const float* __restrict__ PK,     // [NBLK,KVH,BS,D]
    const float* __restrict__ PV,     // [NBLK,KVH,BS,D]
    const int*   __restrict__ SeqPos, // [B]
    const float* __restrict__ Scale,  // [1]
    const int*   __restrict__ BT,     // [B,MAXB]
    float* __restrict__ Out)          // [B,1,H*D]
{
    constexpr int H = 32, KVH = 8, G = 4, D = 128, BS = 128, MAXB = 32;

    __shared__ __align__(16) float q_smem[16 * 128]; // rows 0..3 = rope'd q, 4..15 = 0
    __shared__ __align__(16) float kn[128];          // rope'd new key
    __shared__ __align__(16) float vn[128];          // new value
    __shared__ float red_m[8][4];
    __shared__ float red_l[8][4];
    __shared__ __align__(16) float red_acc[8][4][128];

    const int tid = threadIdx.x;
    const int wg  = blockIdx.x;
    const int b   = wg / KVH;
    const int kvh = wg % KVH;
    const int pos = SeqPos[b];
    const float sc = Scale[0];

    // ---- RoPE query group into LDS (rows 0..3), zero pad rows 4..15
    for (int i = tid; i < 16 * 128; i += 256) {
        int g = i >> 7, d = i & 127;
        float val = 0.0f;
        if (g < G) {
            const float* qb = Q + ((size_t)b * H + (kvh * G + g)) * D;
            float x = qb[d];
            float r = (d < 64) ? -qb[d + 64] : qb[d - 64];
            val = x * Cos[(size_t)b * D + d] + r * Sin[(size_t)b * D + d];
        }
        q_smem[i] = val;
    }
    // ---- RoPE new key / copy new value
    if (tid < 128) {
        int d = tid;
        const float* kb = Knew + ((size_t)b * KVH + kvh) * D;
        float x = kb[d];
        float r = (d < 64) ? -kb[d + 64] : kb[d - 64];
        kn[d] = x * Cos[(size_t)b * D + d] + r * Sin[(size_t)b * D + d];
    } else {
        int d = tid - 128;
        vn[d] = Vnew[((size_t)b * KVH + kvh) * D + d];
    }
    __syncthreads();

    const int lane = tid & 31;
    const int wave = tid >> 5;
    const int lr   = lane & 15;  // row/col within 16
    const int half = lane >> 4;  // K-split half per WMMA operand layout

    float m_run[4] = {NEG_BIG, NEG_BIG, NEG_BIG, NEG_BIG};
    float l_run[4] = {0.f, 0.f, 0.f, 0.f};
    float acc[4][4] = {}; // [group][dd], lane owns d = lane*4 + dd

    const int ntiles = (pos + 15) >> 4; // cache tokens are 0..pos-1
    // A-operand base: q_smem[lr][half*2 + ...]  (16x4 f32 A layout)
    const float* aBase = q_smem + lr * 128 + half * 2;

    for (int tile = wave; tile < ntiles; tile += 8) {
        const int blkidx = BT[b * MAXB + (tile >> 3)];
        const int trow   = (tile & 7) * 16; // 16-token tiles never cross a 128-token block
        const size_t rowbase = ((size_t)blkidx * KVH + kvh) * BS + trow;
        const float* krow = PK + (rowbase + lr) * D + half * 2;

        // ---- prefetch next tile's K/V rows (global_prefetch_b8)
        const int ntile = tile + 8;
        if (ntile < ntiles) {
            const int nblk  = BT[b * MAXB + (ntile >> 3)];
            const int ntrow = (ntile & 7) * 16;
            const size_t nrowbase = ((size_t)nblk * KVH + kvh) * BS + ntrow;
            // lanes 0..31 cover the 16 rows x 2 halves (256B apart) of next K tile
            __builtin_prefetch(PK + (nrowbase + lr) * D + half * 64, 0, 1);
            __builtin_prefetch(PV + (nrowbase + lr) * D + half * 64, 0, 1);
        }

        // ---- scores C[g][tok] = sum_d Q[g][d]*K[tok][d]; 32 chunks of K=4,
        //      split over 4 independent WMMA accumulator chains
        v8f c0 = {}, c1 = {}, c2 = {}, c3 = {};
        #pragma unroll
        for (int ck = 0; ck < 32; ck += 4) {
            v2f a0 = *(const v2f*)(aBase + (ck + 0) * 4);
            v2f b0 = *(const v2f*)(krow + (ck + 0) * 4);
            c0 = __builtin_amdgcn_wmma_f32_16x16x4_f32(
                     false, a0, false, b0, (short)0, c0, false, false);
            v2f a1 = *(const v2f*)(aBase + (ck + 1) * 4);
            v2f b1 = *(const v2f*)(krow + (ck + 1) * 4);
            c1 = __builtin_amdgcn_wmma_f32_16x16x4_f32(
                     false, a1, false, b1, (short)0, c1, false, false);
            v2f a2 = *(const v2f*)(aBase + (ck + 2) * 4);
            v2f b2 = *(const v2f*)(krow + (ck + 2) * 4);
            c2 = __builtin_amdgcn_wmma_f32_16x16x4_f32(
                     false, a2, false, b2, (short)0, c2, false, false);
            v2f a3 = *(const v2f*)(aBase + (ck + 3) * 4);
            v2f b3 = *(const v2f*)(krow + (ck + 3) * 4);
            c3 = __builtin_amdgcn_wmma_f32_16x16x4_f32(
                     false, a3, false, b3, (short)0, c3, false, false);
        }
        v8f c = (c0 + c1) + (c2 + c3);

        const int tok = tile * 16 + lr;              // token for this lane's C column
        const bool inval = (half != 0) || (tok >= pos);
        float p[4];
        #pragma unroll
        for (int g = 0; g < 4; ++g) {
            float s = inval ? NEG_BIG : c[g] * sc;
            // tile max over the 16 valid lanes, broadcast to whole wave
            float r = s;
            #pragma unroll
            for (int mk = 8; mk >= 1; mk >>= 1) r = fmaxf(r, __shfl_xor(r, mk, 32));
            r = __shfl(r, 0, 32);
            float mn    = fmaxf(m_run[g], r);
            float alpha = __expf(m_run[g] - mn);
            m_run[g]    = mn;
            p[g]        = __expf(s - mn); // masked lanes -> 0
            float rs = p[g];
            #pragma unroll
            for (int mk = 8; mk >= 1; mk >>= 1) rs += __shfl_xor(rs, mk, 32);
            rs = __shfl(rs, 0, 32);
            l_run[g] = l_run[g] * alpha + rs;
            #pragma unroll
            for (int dd = 0; dd < 4; ++dd) acc[g][dd] *= alpha;
        }

        // ---- PV: lane owns d = lane*4..lane*4+3; coalesced b128 V-row loads
        const float* vrow = PV + rowbase * D + lane * 4;
        #pragma unroll
        for (int t = 0; t < 16; ++t) {
            float4 vv = *(const float4*)(vrow + t * D);
            float p0 = __shfl(p[0], t, 32);
            float p1 = __shfl(p[1], t, 32);
            float p2 = __shfl(p[2], t, 32);
            float p3 = __shfl(p[3], t, 32);
            acc[0][0] += p0 * vv.x; acc[0][1] += p0 * vv.y;
            acc[0][2] += p0 * vv.z; acc[0][3] += p0 * vv.w;
            acc[1][0] += p1 * vv.x; acc[1][1] += p1 * vv.y;
            acc[1][2] += p1 * vv.z; acc[1][3] += p1 * vv.w;
            acc[2][0] += p2 * vv.x; acc[2][1] += p2 * vv.y;
            acc[2][2] += p2 * vv.z; acc[2][3] += p2 * vv.w;
            acc[3][0] += p3 * vv.x; acc[3][1] += p3 * vv.y;
            acc[3][2] += p3 * vv.z; acc[3][3] += p3 * vv.w;
        }
    }

    // ---- new token at position `pos` (wave 0 only)
    if (wave == 0) {
        #pragma unroll
        for (int g = 0; g < 4; ++g) {
            float partial = 0.f;
            for (int d = lane; d < 128; d += 32)
                partial += q_smem[g * 128 + d] * kn[d];
            #pragma unroll
            for (int mk = 16; mk >= 1; mk >>= 1)
                partial += __shfl_xor(partial, mk, 32);
            float s     = partial * sc;
            float mn    = fmaxf(m_run[g], s);
            float alpha = __expf(m_run[g] - mn);
            float pn    = __expf(s - mn);
            m_run[g] = mn;
            l_run[g] = l_run[g] * alpha + pn;
            #pragma unroll
            for (int dd = 0; dd < 4; ++dd)
                acc[g][dd] = acc[g][dd] * alpha + pn * vn[lane * 4 + dd];
        }
    }

    // ---- publish per-wave partials
    if (lane == 0) {
        #pragma unroll
        for (int g = 0; g < 4; ++g) {
            red_m[wave][g] = m_run[g];
            red_l[wave][g] = l_run[g];
        }
    }
    #pragma unroll
    for (int g = 0; g < 4; ++g) {
        float4 a4 = make_float4(acc[g][0], acc[g][1], acc[g][2], acc[g][3]);
        *(float4*)&red_acc[wave][g][lane * 4] = a4;
    }
    __syncthreads();

    // ---- cross-wave log-sum-exp combine; 512 outputs per workgroup
    for (int o = tid; o < G * D; o += 256) {
        int g = o >> 7, d = o & 127;
        float mstar = NEG_BIG;
        #pragma unroll
        for (int w = 0; w < 8; ++w) mstar = fmaxf(mstar, red_m[w][g]);
        float L = 0.f, val = 0.f;
        #pragma unroll
        for (int w = 0; w < 8; ++w) {
            float f = __expf(red_m[w][g] - mstar);
            L   += f * red_l[w][g];
            val += f * red_acc[w][g][d];
        }
        Out[((size_t)b * H + kvh * G + g) * D + d] = val / L;
    }
}

extern "C" void kernel_launch(void* const* d_in, const int* in_sizes, int n_in,
                              void* d_out, int out_size, void* d_ws, size_t ws_size,
                              hipStream_t stream) {
    const float* q     = (const float*)d_in[0];
    const float* k     = (const float*)d_in[1];
    const float* v     = (const float*)d_in[2];
    const float* cosp  = (const float*)d_in[3];
    const float* sinp  = (const float*)d_in[4];
    const float* pk    = (const float*)d_in[5];
    const float* pv    = (const float*)d_in[6];
    const int*   spos  = (const int*)d_in[7];
    const float* scale = (const float*)d_in[8];
    const int*   bt    = (const int*)d_in[9];
    float* out = (float*)d_out;

    const int B = 32, KVH = 8;
    dim3 grid(B * KVH);
    dim3 block(256);
    DecoderOnlyAttention_43654047596877_kernel<<<grid, block, 0, stream>>>(
        q, k, v, cosp, sinp, pk, pv, spos, scale, bt, out);
}